// Anchor_82300163326779
// MI455X (gfx1250) — compile-verified
//
#include <hip/hip_runtime.h>
#include <math.h>

// Problem constants (from reference)
#define B_N     16
#define C_IN    4
#define L_IN    300000
#define KS      11
#define LC      (L_IN - KS + 1)          // 299990
#define NTILES  ((LC + 15) / 16)         // 18750
#define LEAK    1.0e-4f
#define EPSN    1.0e-12f

typedef float v2f __attribute__((ext_vector_type(2)));
typedef float v8f __attribute__((ext_vector_type(8)));

// --- ordered-uint encoding for float atomicMax (handles negatives) ---
__device__ __forceinline__ unsigned orderedU(float f) {
  unsigned u = __float_as_uint(f);
  return (u & 0x80000000u) ? ~u : (u | 0x80000000u);
}
__device__ __forceinline__ float unorderedF(unsigned u) {
  u = (u & 0x80000000u) ? (u & 0x7FFFFFFFu) : ~u;
  return __uint_as_float(u);
}

// Workspace layout: [0..47] double S (16 batches x {S0,S1,S2}),
// then 16 float sumabs, then 16 uint maxbits.
__global__ void init_ws(double* __restrict__ S, float* __restrict__ sumabs,
                        unsigned* __restrict__ maxbits) {
  int t = threadIdx.x;
  if (t < 48) S[t] = 0.0;
  if (t < 16) { sumabs[t] = 0.0f; maxbits[t] = 0u; }  // 0u == ordered(-NaN), below all reals
}

// -------- Pass 1: single-channel conv via V_WMMA_F32_16X16X4_F32 --------
// D(16x16) = A(16x4 weights, rows replicated) x B(4x16 data tile), 11 taps
// chained through the accumulator. Row 0 of D = acc[0] on lanes 0..15.
__global__ __launch_bounds__(256) void pass1_conv(
    const float* __restrict__ data, const float* __restrict__ W,
    const float* __restrict__ bias, const int* __restrict__ anchorp,
    float* __restrict__ chan, float* __restrict__ sumabs,
    unsigned* __restrict__ maxbits)
{
  const int lane = threadIdx.x & 31;
  const int wv   = threadIdx.x >> 5;
  const int b    = blockIdx.y;
  const int a    = *anchorp;
  const int c0   = (lane & 16) ? 2 : 0;   // channel pair handled by this lane-half
  const int n    = lane & 15;             // output-position column within tile

  const float* p0 = data + ((size_t)b * C_IN + c0) * L_IN;
  const float* p1 = p0 + L_IN;

  // A fragment per tap: lanes 0-15 carry (Wch0,Wch1), lanes 16-31 (Wch2,Wch3)
  v2f afrag[KS];
#pragma unroll
  for (int t = 0; t < KS; ++t) {
    afrag[t].x = W[(a * C_IN + c0    ) * KS + t];
    afrag[t].y = W[(a * C_IN + c0 + 1) * KS + t];
  }
  const float bb = bias[a];

  float psum = 0.0f;
  float pmax = -3.402823466e38f;

  const int wstride = gridDim.x * (256 / 32);
  for (int tile = blockIdx.x * (256 / 32) + wv; tile < NTILES; tile += wstride) {
    const int base = tile * 16;
    const int pos  = base + n;
    v8f acc = {0.f, 0.f, 0.f, 0.f, 0.f, 0.f, 0.f, 0.f};
#pragma unroll
    for (int t = 0; t < KS; ++t) {
      int idx = pos + t;                    // valid outputs never exceed L_IN-1
      if (idx > L_IN - 1) idx = L_IN - 1;   // clamp for the ragged last tile
      v2f bf;
      bf.x = p0[idx];                       // B row (c0)   at column n
      bf.y = p1[idx];                       // B row (c0+1) at column n
      acc = __builtin_amdgcn_wmma_f32_16x16x4_f32(
          false, afrag[t], false, bf, (short)0, acc, false, false);
    }
    float y = acc[0] + bb;                  // D[0][n] (lanes 0-15); dup on 16-31
    y = (y > 0.0f) ? y : LEAK * y;
    if (lane < 16 && pos < LC) {
      chan[(size_t)b * LC + pos] = y;       // stash pre-normalized y
      psum += fabsf(y);
      pmax  = fmaxf(pmax, y);
    }
  }

  __shared__ float ssum[256];
  __shared__ float smax[256];
  ssum[threadIdx.x] = psum;
  smax[threadIdx.x] = pmax;
  __syncthreads();
  for (int s = 128; s > 0; s >>= 1) {
    if (threadIdx.x < (unsigned)s) {
      ssum[threadIdx.x] += ssum[threadIdx.x + s];
      smax[threadIdx.x]  = fmaxf(smax[threadIdx.x], smax[threadIdx.x + s]);
    }
    __syncthreads();
  }
  if (threadIdx.x == 0) {
    atomicAdd(&sumabs[b], ssum[0]);
    atomicMax(&maxbits[b], orderedU(smax[0]));
  }
}

// -------- Pass 2: L1-normalize, write final channel, softmax sums (f64) ----
__global__ __launch_bounds__(256) void pass2_norm(
    float* __restrict__ chan, const float* __restrict__ sumabs,
    const unsigned* __restrict__ maxbits, double* __restrict__ S)
{
  const int b = blockIdx.y;
  const float denom = fmaxf(sumabs[b], EPSN);
  const float cmax  = unorderedF(maxbits[b]) / denom;

  double s0 = 0.0, s1 = 0.0, s2 = 0.0;
  const int stride = gridDim.x * 256;
  for (int i = blockIdx.x * 256 + threadIdx.x; i < LC; i += stride) {
    float c = chan[(size_t)b * LC + i] / denom;
    chan[(size_t)b * LC + i] = c;           // final `channel` output
    float e = expf(c - cmax);
    double x = (double)i;
    s0 += (double)e;
    s1 += (double)e * x;
    s2 += (double)e * x * x;
  }

  __shared__ double sh0[256], sh1[256], sh2[256];
  sh0[threadIdx.x] = s0; sh1[threadIdx.x] = s1; sh2[threadIdx.x] = s2;
  __syncthreads();
  for (int s = 128; s > 0; s >>= 1) {
    if (threadIdx.x < (unsigned)s) {
      sh0[threadIdx.x] += sh0[threadIdx.x + s];
      sh1[threadIdx.x] += sh1[threadIdx.x + s];
      sh2[threadIdx.x] += sh2[threadIdx.x + s];
    }
    __syncthreads();
  }
  if (threadIdx.x == 0) {
    atomicAdd(&S[b * 3 + 0], sh0[0]);
    atomicAdd(&S[b * 3 + 1], sh1[0]);
    atomicAdd(&S[b * 3 + 2], sh2[0]);
  }
}

// -------- Pass 3: position-only z-moments (no memory traffic) -------------
__global__ __launch_bounds__(256) void pass3_moments(
    const double* __restrict__ S, float* __restrict__ skews,
    float* __restrict__ kts)
{
  const int b = blockIdx.x;
  const double S0 = S[b * 3 + 0];
  const double S1 = S[b * 3 + 1];
  const double S2 = S[b * 3 + 2];
  const double mean = S1 / S0;
  const double var  = S2 / S0 - mean * mean;
  const double inv  = 1.0 / sqrt(var);

  double z3 = 0.0, z4 = 0.0;
  for (int i = threadIdx.x; i < LC; i += 256) {
    double z  = ((double)i - mean) * inv;
    double c3 = z * z * z;
    z3 += c3;
    z4 += c3 * z;
  }
  __shared__ double sh3[256], sh4[256];
  sh3[threadIdx.x] = z3; sh4[threadIdx.x] = z4;
  __syncthreads();
  for (int s = 128; s > 0; s >>= 1) {
    if (threadIdx.x < (unsigned)s) {
      sh3[threadIdx.x] += sh3[threadIdx.x + s];
      sh4[threadIdx.x] += sh4[threadIdx.x + s];
    }
    __syncthreads();
  }
  if (threadIdx.x == 0) {
    skews[b] = (float)(sh3[0] / (double)LC);
    kts[b]   = (float)(sh4[0] / (double)LC - 3.0);
  }
}

extern "C" void kernel_launch(void* const* d_in, const int* in_sizes, int n_in,
                              void* d_out, int out_size, void* d_ws, size_t ws_size,
                              hipStream_t stream) {
  const float* data    = (const float*)d_in[0];  // [16,4,300000]
  const float* W       = (const float*)d_in[1];  // [16,4,11]
  const float* bias    = (const float*)d_in[2];  // [16]
  const int*   anchorp = (const int*)d_in[3];    // scalar

  float* out   = (float*)d_out;
  float* skews = out;          // [16]
  float* kts   = out + 16;     // [16]
  float* chan  = out + 32;     // [16, 299990]

  double*   S       = (double*)d_ws;             // 48 doubles
  float*    sumabs  = (float*)(S + 48);          // 16 floats
  unsigned* maxbits = (unsigned*)(sumabs + 16);  // 16 uints

  init_ws<<<1, 64, 0, stream>>>(S, sumabs, maxbits);
  pass1_conv<<<dim3(64, B_N), 256, 0, stream>>>(data, W, bias, anchorp,
                                                chan, sumabs, maxbits);
  pass2_norm<<<dim3(128, B_N), 256, 0, stream>>>(chan, sumabs, maxbits, S);
  pass3_moments<<<B_N, 256, 0, stream>>>(S, skews, kts);
}